// VarianceAdaptor_38087769981325
// MI455X (gfx1250) — compile-verified
//
#include <hip/hip_runtime.h>
#include <hip/hip_bf16.h>

typedef float v2f __attribute__((ext_vector_type(2)));
typedef float v8f __attribute__((ext_vector_type(8)));

constexpr int Bn = 32;
constexpr int Cc = 384;
constexpr int Tp = 512;
constexpr int Tf = 2048;

// ---------------------------------------------------------------------------
// K1: per-batch inclusive scan of duration + binary search: frame -> phone idx
// attn_path[b,p,f] == 1  iff  cum[p-1] <= f < cum[p]  (and masks nonzero)
// ---------------------------------------------------------------------------
__global__ __launch_bounds__(Tp) void k_path(const int* __restrict__ dur,
                                             const float* __restrict__ pmask,
                                             const float* __restrict__ fmask,
                                             int* __restrict__ idx) {
  __shared__ int s[Tp];
  const int b = blockIdx.x;
  const int t = threadIdx.x;
  s[t] = dur[b * Tp + t];
  __syncthreads();
  // Hillis-Steele inclusive scan
  for (int off = 1; off < Tp; off <<= 1) {
    int v = (t >= off) ? s[t - off] : 0;
    __syncthreads();
    s[t] += v;
    __syncthreads();
  }
  const int total = s[Tp - 1];
  for (int f = t; f < Tf; f += Tp) {
    int p = -1;
    if (f < total) {
      int lo = 0, hi = Tp - 1;
      while (lo < hi) {           // smallest p with cum[p] > f
        int mid = (lo + hi) >> 1;
        if (s[mid] > f) hi = mid; else lo = mid + 1;
      }
      p = lo;
      if (pmask[b * Tp + p] == 0.f || fmask[b * Tf + f] == 0.f) p = -1;
    }
    idx[b * Tf + f] = p;
  }
}

// ---------------------------------------------------------------------------
// K2: fused gather (xf) + rank-1 embeddings + out store + 3-row projection
//     via V_WMMA_F32_16X16X4_F32 (M rows: pitch0, pitch1, energy; rows 3..15
//     zero-padded in LDS so the inner loop is completely branch-free).
// One wave = 16 frames. Block = 8 waves = 128 frames. Grid = 32 b * 16 tiles.
// ---------------------------------------------------------------------------
__global__ __launch_bounds__(256) void k_main(
    const float* __restrict__ x, const float* __restrict__ fmask,
    const int* __restrict__ idx, const float* __restrict__ lcf0,
    const float* __restrict__ energy, const float* __restrict__ pitch_w,
    const float* __restrict__ pitch_b, const float* __restrict__ pemb_w,
    const float* __restrict__ pemb_b, const float* __restrict__ enp_w,
    const float* __restrict__ enp_b, const float* __restrict__ eemb_w,
    const float* __restrict__ eemb_b, float* __restrict__ out,
    float* __restrict__ lcf0_pred, float* __restrict__ vuv_pred,
    float* __restrict__ en_pred) {
  __shared__ float WA[16 * Cc];  // 16 WMMA rows: pitch0, pitch1, enp, 13x zero
  __shared__ float PW[Cc], PB[Cc], EW[Cc], EB[Cc];
  const int tid = threadIdx.x;
  // A-matrix: rows 0-1 = pitch_w (flat 2*Cc), row 2 = enp_w, rows 3-15 = 0
  for (int i = tid; i < 16 * Cc; i += 256) {
    float v = 0.f;
    if (i < 2 * Cc)      v = pitch_w[i];
    else if (i < 3 * Cc) v = enp_w[i - 2 * Cc];
    WA[i] = v;
  }
  for (int i = tid; i < 4 * Cc; i += 256) {
    const int seg = i / Cc, c = i % Cc;
    if (seg == 0)      PW[c] = pemb_w[c];
    else if (seg == 1) PB[c] = pemb_b[c];
    else if (seg == 2) EW[c] = eemb_w[c];
    else               EB[c] = eemb_b[c];
  }
  __syncthreads();

  const int bx = blockIdx.x;
  const int b = bx >> 4;                  // 16 tiles per batch
  const int fblock = (bx & 15) * 128;
  const int wave = tid >> 5, lane = tid & 31;
  const int col = lane & 15;              // A: row M;  B: frame column N
  const int half = lane >> 4;             // K selector (0 -> K0/K1, 1 -> K2/K3)
  const int f = fblock + wave * 16 + col;
  const size_t fo = (size_t)b * Tf + f;

  const int p = idx[fo];
  const float valid = (p >= 0) ? 1.f : 0.f;  // branch-free gather mask
  const int pc = (p >= 0) ? p : 0;           // always-legal address
  const float fm = fmask[fo];
  const float l0 = lcf0[fo];
  const float en = energy[fo];
  const float* xb = x + (size_t)b * Cc * Tp;
  float* ob = out + (size_t)b * Cc * Tf;

  v8f acc = {};
  for (int k = 0; k < Cc / 4; ++k) {
    const int c0 = 4 * k + 2 * half;
    // A operand: unconditional LDS load (rows >= 3 read zeros)
    const float a0 = WA[col * Cc + c0];
    const float a1 = WA[col * Cc + c0 + 1];
    // B operand: unconditional gather, zeroed by `valid` multiply
    const float b0 = xb[(size_t)c0 * Tp + pc] * valid;
    const float b1 = xb[(size_t)(c0 + 1) * Tp + pc] * valid;
    v2f A = {a0, a1};
    v2f Bv = {b0, b1};
    acc = __builtin_amdgcn_wmma_f32_16x16x4_f32(
        /*neg_a=*/false, A, /*neg_b=*/false, Bv,
        /*c_mod=*/(short)0, acc, /*reuse_a=*/false, /*reuse_b=*/false);
    // out = xf + fm*(lcf0*pemb_w + pemb_b) + fm*(energy*eemb_w + eemb_b)
    const float e0 = fm * (l0 * PW[c0]     + PB[c0]     + en * EW[c0]     + EB[c0]);
    const float e1 = fm * (l0 * PW[c0 + 1] + PB[c0 + 1] + en * EW[c0 + 1] + EB[c0 + 1]);
    ob[(size_t)c0 * Tf + f]       = b0 + e0;
    ob[(size_t)(c0 + 1) * Tf + f] = b1 + e1;
  }
  // C/D layout: VGPR j, lanes 0-15 hold row M=j, N=lane
  if (lane < 16) {
    lcf0_pred[fo] = (acc[0] + pitch_b[0]) * fm;  // row 0 = pitch_w[0]
    vuv_pred[fo]  = (acc[1] + pitch_b[1]) * fm;  // row 1 = pitch_w[1]
    en_pred[fo]   = (acc[2] + enp_b[0])  * fm;   // row 2 = enp_w
  }
}

// ---------------------------------------------------------------------------
// K3: duration head: dot over C per (b, t), coalesced across t.
// ---------------------------------------------------------------------------
__global__ __launch_bounds__(256) void k_dur(const float* __restrict__ x,
                                             const float* __restrict__ pmask,
                                             const float* __restrict__ dur_w,
                                             const float* __restrict__ dur_b,
                                             float* __restrict__ dur_pred) {
  const int g = blockIdx.x * 256 + threadIdx.x;  // B*Tp threads
  const int b = g / Tp, t = g % Tp;
  const float* xb = x + (size_t)b * Cc * Tp + t;
  float acc = 0.f;
#pragma unroll 4
  for (int c = 0; c < Cc; ++c) acc += xb[(size_t)c * Tp] * dur_w[c];
  dur_pred[g] = (acc + dur_b[0]) * pmask[g];
}

extern "C" void kernel_launch(void* const* d_in, const int* in_sizes, int n_in,
                              void* d_out, int out_size, void* d_ws,
                              size_t ws_size, hipStream_t stream) {
  const float* x       = (const float*)d_in[0];
  const float* pmask   = (const float*)d_in[1];
  const float* fmask   = (const float*)d_in[2];
  const int*   dur     = (const int*)d_in[3];
  const float* lcf0    = (const float*)d_in[4];
  // d_in[5] = vuv (unused by reference)
  const float* energy  = (const float*)d_in[6];
  const float* dur_w   = (const float*)d_in[7];
  const float* dur_b   = (const float*)d_in[8];
  const float* pitch_w = (const float*)d_in[9];
  const float* pitch_b = (const float*)d_in[10];
  const float* pemb_w  = (const float*)d_in[11];
  const float* pemb_b  = (const float*)d_in[12];
  const float* enp_w   = (const float*)d_in[13];
  const float* enp_b   = (const float*)d_in[14];
  const float* eemb_w  = (const float*)d_in[15];
  const float* eemb_b  = (const float*)d_in[16];

  float* out       = (float*)d_out;                      // (B, C, Tf)
  float* dur_pred  = out + (size_t)Bn * Cc * Tf;         // (B, 1, Tp)
  float* lcf0_pred = dur_pred + (size_t)Bn * Tp;         // (B, 1, Tf)
  float* vuv_pred  = lcf0_pred + (size_t)Bn * Tf;        // (B, 1, Tf)
  float* en_pred   = vuv_pred + (size_t)Bn * Tf;         // (B, 1, Tf)

  int* idx = (int*)d_ws;  // B*Tf ints = 256 KB

  k_path<<<Bn, Tp, 0, stream>>>(dur, pmask, fmask, idx);
  k_main<<<Bn * 16, 256, 0, stream>>>(x, fmask, idx, lcf0, energy, pitch_w,
                                      pitch_b, pemb_w, pemb_b, enp_w, enp_b,
                                      eemb_w, eemb_b, out, lcf0_pred, vuv_pred,
                                      en_pred);
  k_dur<<<(Bn * Tp) / 256, 256, 0, stream>>>(x, pmask, dur_w, dur_b, dur_pred);
}